// GraphSAGE_5454608466411
// MI455X (gfx1250) — compile-verified
//
#include <hip/hip_runtime.h>
#include <hip/hip_bf16.h>

typedef __attribute__((ext_vector_type(2))) float v2f;
typedef __attribute__((ext_vector_type(8))) float v8f;

// ---------------------------------------------------------------------------
// Utility kernels
// ---------------------------------------------------------------------------
__global__ void zero_kernel(float* __restrict__ p, int n) {
    int i = blockIdx.x * blockDim.x + threadIdx.x;
    if (i < n) p[i] = 0.0f;
}

__global__ void degree_kernel(const int* __restrict__ dst, float* __restrict__ deg, int E) {
    int e = blockIdx.x * blockDim.x + threadIdx.x;
    if (e < E) atomicAdd(&deg[dst[e]], 1.0f);
}

__global__ void invdeg_kernel(const float* __restrict__ deg, float* __restrict__ invd, int N) {
    int n = blockIdx.x * blockDim.x + threadIdx.x;
    if (n < N) invd[n] = 1.0f / fmaxf(deg[n], 1.0f);
}

// ---------------------------------------------------------------------------
// Edge scatter-add: agg[dst] += x[src]; one thread handles 4 consecutive feats
// ---------------------------------------------------------------------------
template<int D>
__global__ void scatter_add_kernel(const float* __restrict__ xin,
                                   const int* __restrict__ src,
                                   const int* __restrict__ dst,
                                   float* __restrict__ agg, int E) {
    const int Q = D / 4;
    int idx = blockIdx.x * blockDim.x + threadIdx.x;
    if (idx >= E * Q) return;
    int e = idx / Q;
    int q = idx - e * Q;
    int s = src[e];
    int d = dst[e];
    const float4* xs = (const float4*)(xin + (size_t)s * D);
    float4 v = xs[q];
    float* ap = agg + (size_t)d * D + q * 4;
    atomicAdd(ap + 0, v.x);
    atomicAdd(ap + 1, v.y);
    atomicAdd(ap + 2, v.z);
    atomicAdd(ap + 3, v.w);
}

// ---------------------------------------------------------------------------
// Fused SAGE layer GEMM: hpre = (agg*invdeg) @ Wl + hself @ Wr + b
// One wave32 per 16x16 output tile, f32 WMMA (16x16x4), K in {128, 64}, H=64.
// Also accumulates per-feature BN partial sums (sum, sumsq) via atomics.
// ---------------------------------------------------------------------------
template<int K>
__global__ void sage_gemm_kernel(const float* __restrict__ Aagg,
                                 const float* __restrict__ Aself,
                                 const float* __restrict__ invd,
                                 const float* __restrict__ Wl,
                                 const float* __restrict__ Wr,
                                 const float* __restrict__ bias,
                                 float* __restrict__ hpre,
                                 float* __restrict__ sums,
                                 float* __restrict__ sumsq,
                                 int n_row_tiles) {
    const int H = 64;
    int wave = blockIdx.x * (blockDim.x >> 5) + (threadIdx.x >> 5);
    int row_tile = wave >> 2;       // 4 col tiles (H=64)
    int col_tile = wave & 3;
    if (row_tile >= n_row_tiles) return;   // wave-uniform exit, EXEC stays full

    int lane = threadIdx.x & 31;
    int half = lane >> 4;           // selects K-pair within 4-wide K step
    int l    = lane & 15;
    int row  = row_tile * 16 + l;   // A-frag row (M)
    int col  = col_tile * 16 + l;   // B-frag col (N)

    float scale = invd[row];
    const float* arow = Aagg  + (size_t)row * K;
    const float* srow = Aself + (size_t)row * K;

    v8f c = {0.f, 0.f, 0.f, 0.f, 0.f, 0.f, 0.f, 0.f};

    #pragma unroll 4
    for (int k0 = 0; k0 < K; k0 += 4) {
        int ka = k0 + half * 2;
        v2f am, as, bl, br;
        am.x = arow[ka] * scale;  am.y = arow[ka + 1] * scale;
        as.x = srow[ka];          as.y = srow[ka + 1];
        bl.x = Wl[ka * H + col];  bl.y = Wl[(ka + 1) * H + col];
        br.x = Wr[ka * H + col];  br.y = Wr[(ka + 1) * H + col];
        c = __builtin_amdgcn_wmma_f32_16x16x4_f32(false, am, false, bl,
                                                  (short)0, c, false, false);
        c = __builtin_amdgcn_wmma_f32_16x16x4_f32(false, as, false, br,
                                                  (short)0, c, false, false);
    }

    float bj = bias[col];
    float s = 0.f, sq = 0.f;
    #pragma unroll
    for (int i = 0; i < 8; ++i) {
        float v = c[i] + bj;
        int r = row_tile * 16 + half * 8 + i;   // C/D layout: M = i + 8*half
        hpre[(size_t)r * H + col] = v;
        s  += v;
        sq += v * v;
    }
    atomicAdd(&sums[col],  s);
    atomicAdd(&sumsq[col], sq);
}

// ---------------------------------------------------------------------------
// BN finalize: sums/sumsq (64 each) -> scale/shift in place
// ---------------------------------------------------------------------------
__global__ void bn_finalize_kernel(float* __restrict__ sums, float* __restrict__ sumsq,
                                   const float* __restrict__ gamma,
                                   const float* __restrict__ beta, float invN) {
    int j = threadIdx.x;            // 64 threads
    float mu  = sums[j] * invN;
    float var = sumsq[j] * invN - mu * mu;
    float rs  = rsqrtf(var + 1e-5f);
    float sc  = rs * gamma[j];
    sums[j]  = sc;                  // scale
    sumsq[j] = beta[j] - mu * sc;   // shift
}

// ---------------------------------------------------------------------------
// Elementwise BN apply + ReLU (float4 per thread, H=64)
// ---------------------------------------------------------------------------
__global__ void bn_relu_kernel(const float* __restrict__ hpre,
                               const float* __restrict__ scale,
                               const float* __restrict__ shift,
                               float* __restrict__ hout, int total4) {
    int i = blockIdx.x * blockDim.x + threadIdx.x;
    if (i >= total4) return;
    float4 v = ((const float4*)hpre)[i];
    int j = (i * 4) & 63;
    v.x = fmaxf(v.x * scale[j + 0] + shift[j + 0], 0.f);
    v.y = fmaxf(v.y * scale[j + 1] + shift[j + 1], 0.f);
    v.z = fmaxf(v.z * scale[j + 2] + shift[j + 2], 0.f);
    v.w = fmaxf(v.w * scale[j + 3] + shift[j + 3], 0.f);
    ((float4*)hout)[i] = v;
}

// ---------------------------------------------------------------------------
// Output head: out[n] = h[n,:] . Wout + bout ; one wave32 per node
// ---------------------------------------------------------------------------
__global__ void out_kernel(const float* __restrict__ h,
                           const float* __restrict__ Wout,
                           const float* __restrict__ bout,
                           float* __restrict__ out, int N) {
    int wave = blockIdx.x * (blockDim.x >> 5) + (threadIdx.x >> 5);
    int lane = threadIdx.x & 31;
    if (wave >= N) return;
    const float* hr = h + (size_t)wave * 64;
    float acc = hr[lane] * Wout[lane] + hr[lane + 32] * Wout[lane + 32];
    #pragma unroll
    for (int off = 16; off > 0; off >>= 1)
        acc += __shfl_down(acc, off, 32);
    if (lane == 0) out[wave] = acc + bout[0];
}

// ---------------------------------------------------------------------------
// Launch
// ---------------------------------------------------------------------------
extern "C" void kernel_launch(void* const* d_in, const int* in_sizes, int n_in,
                              void* d_out, int out_size, void* d_ws, size_t ws_size,
                              hipStream_t stream) {
    const float* x    = (const float*)d_in[0];
    const int*   edge = (const int*)  d_in[1];
    const float* Wl0  = (const float*)d_in[2];
    const float* Wr0  = (const float*)d_in[3];
    const float* b0   = (const float*)d_in[4];
    const float* Wl1  = (const float*)d_in[5];
    const float* Wr1  = (const float*)d_in[6];
    const float* b1   = (const float*)d_in[7];
    const float* Wl2  = (const float*)d_in[8];
    const float* Wr2  = (const float*)d_in[9];
    const float* b2   = (const float*)d_in[10];
    const float* g0   = (const float*)d_in[11];
    const float* be0  = (const float*)d_in[12];
    const float* g1   = (const float*)d_in[13];
    const float* be1  = (const float*)d_in[14];
    const float* g2   = (const float*)d_in[15];
    const float* be2  = (const float*)d_in[16];
    const float* Wout = (const float*)d_in[17];
    const float* bout = (const float*)d_in[18];
    float* out = (float*)d_out;

    const int N = in_sizes[0] / 128;
    const int E = in_sizes[1] / 2;
    const int* src = edge;
    const int* dst = edge + E;

    float* ws    = (float*)d_ws;
    float* deg   = ws;                 ws += N;
    float* invd  = ws;                 ws += N;
    float* sums  = ws;                 ws += 64;
    float* sumsq = ws;                 ws += 64;
    float* agg   = ws;                 ws += (size_t)N * 128;
    float* hpre  = ws;                 ws += (size_t)N * 64;
    float* hA    = ws;                 ws += (size_t)N * 64;
    float* hB    = ws;                 ws += (size_t)N * 64;

    const int n_row_tiles = (N + 15) / 16;       // N=50000 -> 3125 exact
    const int gemm_blocks = (n_row_tiles * 4 + 7) / 8;   // 8 waves / block
    const float invN = 1.0f / (float)N;

    #define ZERO(p, n) zero_kernel<<<((n) + 255) / 256, 256, 0, stream>>>((p), (int)(n))

    // degrees (shared by all layers)
    ZERO(deg, N);
    degree_kernel<<<(E + 255) / 256, 256, 0, stream>>>(dst, deg, E);
    invdeg_kernel<<<(N + 255) / 256, 256, 0, stream>>>(deg, invd, N);

    // -------- layer 0 (K = 128) --------
    ZERO(agg, (size_t)N * 128); ZERO(sums, 64); ZERO(sumsq, 64);
    scatter_add_kernel<128><<<(unsigned)(((size_t)E * 32 + 255) / 256), 256, 0, stream>>>(x, src, dst, agg, E);
    sage_gemm_kernel<128><<<gemm_blocks, 256, 0, stream>>>(agg, x, invd, Wl0, Wr0, b0,
                                                           hpre, sums, sumsq, n_row_tiles);
    bn_finalize_kernel<<<1, 64, 0, stream>>>(sums, sumsq, g0, be0, invN);
    bn_relu_kernel<<<((N * 16) + 255) / 256, 256, 0, stream>>>(hpre, sums, sumsq, hA, N * 16);

    // -------- layer 1 (K = 64) --------
    ZERO(agg, (size_t)N * 64); ZERO(sums, 64); ZERO(sumsq, 64);
    scatter_add_kernel<64><<<(unsigned)(((size_t)E * 16 + 255) / 256), 256, 0, stream>>>(hA, src, dst, agg, E);
    sage_gemm_kernel<64><<<gemm_blocks, 256, 0, stream>>>(agg, hA, invd, Wl1, Wr1, b1,
                                                          hpre, sums, sumsq, n_row_tiles);
    bn_finalize_kernel<<<1, 64, 0, stream>>>(sums, sumsq, g1, be1, invN);
    bn_relu_kernel<<<((N * 16) + 255) / 256, 256, 0, stream>>>(hpre, sums, sumsq, hB, N * 16);

    // -------- layer 2 (K = 64) --------
    ZERO(agg, (size_t)N * 64); ZERO(sums, 64); ZERO(sumsq, 64);
    scatter_add_kernel<64><<<(unsigned)(((size_t)E * 16 + 255) / 256), 256, 0, stream>>>(hB, src, dst, agg, E);
    sage_gemm_kernel<64><<<gemm_blocks, 256, 0, stream>>>(agg, hB, invd, Wl2, Wr2, b2,
                                                          hpre, sums, sumsq, n_row_tiles);
    bn_finalize_kernel<<<1, 64, 0, stream>>>(sums, sumsq, g2, be2, invN);
    bn_relu_kernel<<<((N * 16) + 255) / 256, 256, 0, stream>>>(hpre, sums, sumsq, hA, N * 16);

    // -------- output head --------
    out_kernel<<<(N + 7) / 8, 256, 0, stream>>>(hA, Wout, bout, out, N);

    #undef ZERO
}